// RandBox_89919435309523
// MI455X (gfx1250) — compile-verified
//
#include <hip/hip_runtime.h>
#include <hip/hip_bf16.h>
#include <cstdint>

#define B_        4
#define N_        2000
#define MAXF_     50
#define H_        1024
#define W_        1024
#define NPAD      2048
#define NMS_THR_  0.7f
#define MIN_SCALE_ 0.1f

// ---------------------------------------------------------------------------
// CDNA5 async global<->LDS DMA helpers (ASYNCcnt-tracked, gfx1250-only path).
// global_load_async_to_lds_b32 : copies 4 bytes per lane from the global address in vaddr to the LDS address in vdst
// global_store_async_from_lds_b128 : copies 16 bytes per lane from the LDS address in vsrc to the global address in vaddr
// s_wait_asynccnt 0 : required between them (loads/stores retire unordered
// w.r.t. each other on ASYNCcnt per CDNA5 ISA §7.1).
// ---------------------------------------------------------------------------
__device__ __forceinline__ void async_load_b32_to_lds(unsigned lds_addr, const void* gptr) {
  asm volatile("global_load_async_to_lds_b32 %0, %1, off"
               :: "v"(lds_addr), "v"((unsigned long long)(size_t)gptr)
               : "memory");
}
__device__ __forceinline__ void async_store_b128_from_lds(unsigned lds_addr, void* gptr) {
  asm volatile("global_store_async_from_lds_b128 %1, %0, off"
               :: "v"(lds_addr), "v"((unsigned long long)(size_t)gptr)
               : "memory");
}
__device__ __forceinline__ void wait_asynccnt0() {
  asm volatile("s_wait_asynccnt 0x0" ::: "memory");
}

// ---------------------------------------------------------------------------
// Kernel 1: img_1 = img[:, :3] (copy), img_2 = flip(img[:, 3:], axis=-1).
// One block per (b, c, y) row. Flip goes entirely through LDS via async DMA:
// the reversal happens in the per-lane DMA addressing, never in VGPRs.
// ---------------------------------------------------------------------------
__global__ __launch_bounds__(256) void copy_flip_kernel(const float* __restrict__ img,
                                                        float* __restrict__ img1,
                                                        float* __restrict__ img2) {
  const int t = threadIdx.x;
  const int r = blockIdx.x;            // [0, B*3*H)
  const int y = r & (H_ - 1);
  const int q = r >> 10;               // H_ == 1024
  const int c = q % 3;
  const int b = q / 3;

  const size_t rowIn1 = (((size_t)b * 6 + c) * H_ + y) * (size_t)W_;
  const size_t rowIn2 = (((size_t)b * 6 + 3 + c) * H_ + y) * (size_t)W_;
  const size_t rowOut = (((size_t)b * 3 + c) * H_ + y) * (size_t)W_;

  // img_1: straight wide copy (global_load_b128 / global_store_b128)
  const float4* s4 = reinterpret_cast<const float4*>(img + rowIn1);
  float4*       d4 = reinterpret_cast<float4*>(img1 + rowOut);
  d4[t] = s4[t];

  // img_2: W-flip, DMA'd through LDS. LDS element e holds input element W-1-e, so the
  // staged row is fully reversed and can be streamed out with b128 DMA.
  __shared__ float rowbuf[W_];
  const unsigned lds0 = (unsigned)(size_t)(void*)rowbuf;  // low 32 bits = LDS offset
  const float* in2 = img + rowIn2;
#pragma unroll
  for (int kk = 0; kk < 4; ++kk) {
    const int e = t + kk * 256;
    async_load_b32_to_lds(lds0 + (unsigned)e * 4u, in2 + (W_ - 1 - e));
  }
  wait_asynccnt0();      // my wave's async loads have landed in LDS
  __syncthreads();       // everyone's have
  async_store_b128_from_lds(lds0 + (unsigned)t * 16u, img2 + rowOut + (size_t)t * 4);
  // implicit s_wait_idle at s_endpgm drains the outstanding async store
}

// ---------------------------------------------------------------------------
// Kernel 2: per-image box pipeline. One block (1024 threads) per image.
// build keys -> bitonic sort (desc) -> greedy NMS -> rank/emit.
// ---------------------------------------------------------------------------
__global__ __launch_bounds__(1024) void nms_kernel(const float* __restrict__ rand_init,
                                                   const float* __restrict__ scores,
                                                   const int*   __restrict__ num_boxes,
                                                   const int*   __restrict__ ph,
                                                   const int*   __restrict__ pw,
                                                   float* __restrict__ out_boxes,
                                                   float* __restrict__ boxes2,
                                                   float* __restrict__ counts) {
  __shared__ float  key[NPAD];          // sorted score key (valid: [0,1), invalid: -1, pad: -2)
  __shared__ int    sid[NPAD];          // payload index, later reused for ranks
  __shared__ float4 sbox[NPAD];         // sorted boxes (x0,y0,x1,y1)
  __shared__ unsigned char supp[NPAD];
  __shared__ unsigned char keepf[NPAD];
  __shared__ int s_count;

  const int b   = blockIdx.x;
  const int tid = threadIdx.x;
  const float Himg = (float)ph[0];
  const float Wimg = (float)pw[0];
  const float* ri = rand_init + (size_t)b * N_ * 4;
  const float* sc = scores    + (size_t)b * N_;

  for (int i = tid; i < NPAD; i += 1024) {
    float k = -2.0f;
    if (i < N_) {
      const float r0 = ri[i * 4 + 0], r1 = ri[i * 4 + 1];
      const float r2 = ri[i * 4 + 2], r3 = ri[i * 4 + 3];
      const float x0 = fminf(r0, r2) * Wimg, x1 = fmaxf(r0, r2) * Wimg;
      const float y0 = fminf(r1, r3) * Himg, y1 = fmaxf(r1, r3) * Himg;
      const bool valid = ((y1 - y0) > Himg * MIN_SCALE_) && ((x1 - x0) > Wimg * MIN_SCALE_);
      k = valid ? sc[i] : -1.0f;
    }
    key[i]   = k;
    sid[i]   = i;
    supp[i]  = 0;
    keepf[i] = 0;
  }
  __syncthreads();

  // bitonic sort: descending by key, payload sid. 2048 = 2^11 -> 66 stages.
  for (int kk = 2; kk <= NPAD; kk <<= 1) {
    for (int j = kk >> 1; j > 0; j >>= 1) {
      for (int i = tid; i < NPAD; i += 1024) {
        const int ixj = i ^ j;
        if (ixj > i) {
          const float a = key[i], c = key[ixj];
          const bool desc = ((i & kk) == 0);
          if (desc ? (a < c) : (a > c)) {
            key[i] = c; key[ixj] = a;
            const int t2 = sid[i]; sid[i] = sid[ixj]; sid[ixj] = t2;
          }
        }
      }
      __syncthreads();
    }
  }

  // gather sorted boxes (recompute from rand_init; cheaper than 2x LDS)
  for (int jj = tid; jj < NPAD; jj += 1024) {
    const int i = sid[jj];
    float4 bx = make_float4(0.f, 0.f, 0.f, 0.f);
    if (i < N_) {
      const float r0 = ri[i * 4 + 0], r1 = ri[i * 4 + 1];
      const float r2 = ri[i * 4 + 2], r3 = ri[i * 4 + 3];
      bx.x = fminf(r0, r2) * Wimg; bx.z = fmaxf(r0, r2) * Wimg;
      bx.y = fminf(r1, r3) * Himg; bx.w = fmaxf(r1, r3) * Himg;
    }
    sbox[jj] = bx;
  }
  __syncthreads();

  // greedy NMS. Decision reads (key[i], supp[i]) are uniform LDS broadcasts,
  // so control flow is workgroup-uniform; skipped pivots write nothing and
  // need no barrier, shortening the sequential chain.
  for (int i = 0; i < N_; ++i) {
    const bool kp = (key[i] >= 0.0f) && (supp[i] == 0);
    if (kp) {
      if (tid == 0) keepf[i] = 1;
      const float4 bi = sbox[i];
      const float areai = (bi.z - bi.x) * (bi.w - bi.y);
      for (int jj = i + 1 + tid; jj < N_; jj += 1024) {
        if (supp[jj] == 0 && key[jj] >= 0.0f) {
          const float4 bj = sbox[jj];
          const float lx = fmaxf(bi.x, bj.x), ly = fmaxf(bi.y, bj.y);
          const float rx = fminf(bi.z, bj.z), ry = fminf(bi.w, bj.w);
          const float w = fmaxf(rx - lx, 0.f), h = fmaxf(ry - ly, 0.f);
          const float inter = w * h;
          const float areaj = (bj.z - bj.x) * (bj.w - bj.y);
          const float iou = inter / (areai + areaj - inter + 1e-12f);
          if (iou > NMS_THR_) supp[jj] = 1;
        }
      }
      __syncthreads();
    }
  }
  __syncthreads();

  // ranks + count (tiny sequential pass, reuse sid[])
  if (tid == 0) {
    int cnt = 0;
    for (int i = 0; i < N_; ++i) {
      sid[i] = keepf[i] ? cnt : -1;
      if (keepf[i]) ++cnt;
    }
    const int nb = num_boxes[b];
    s_count = cnt < nb ? cnt : nb;
  }
  __syncthreads();
  const int count = s_count;

  float* ob = out_boxes + (size_t)b * MAXF_ * 4;
  float* b2 = boxes2    + (size_t)b * MAXF_ * 4;
  for (int i = tid; i < MAXF_ * 4; i += 1024) { ob[i] = 0.0f; b2[i] = 0.0f; }
  __syncthreads();
  for (int i = tid; i < N_; i += 1024) {
    if (keepf[i]) {
      const int r = sid[i];
      if (r >= 0 && r < count) {
        const float4 bx = sbox[i];
        ob[r * 4 + 0] = bx.x; ob[r * 4 + 1] = bx.y;
        ob[r * 4 + 2] = bx.z; ob[r * 4 + 3] = bx.w;
        b2[r * 4 + 0] = Wimg - (bx.z + 1.0f);
        b2[r * 4 + 1] = bx.y;
        b2[r * 4 + 2] = Wimg - (bx.x + 1.0f);
        b2[r * 4 + 3] = bx.w;
      }
    }
  }
  if (tid == 0) counts[b] = (float)count;
}

// ---------------------------------------------------------------------------
// Outputs flat in return order: out_boxes(4,50,4) | boxes_2(4,50,4) |
// counts(4) | img_1(4,3,1024,1024) | img_2(4,3,1024,1024)
// ---------------------------------------------------------------------------
extern "C" void kernel_launch(void* const* d_in, const int* in_sizes, int n_in,
                              void* d_out, int out_size, void* d_ws, size_t ws_size,
                              hipStream_t stream) {
  const float* img       = (const float*)d_in[0];
  const float* rand_init = (const float*)d_in[1];
  const float* scores    = (const float*)d_in[2];
  const int*   num_boxes = (const int*)d_in[3];
  const int*   ph        = (const int*)d_in[4];
  const int*   pw        = (const int*)d_in[5];

  float* out       = (float*)d_out;
  float* out_boxes = out;
  float* boxes2    = out + (size_t)B_ * MAXF_ * 4;
  float* counts    = out + (size_t)2 * B_ * MAXF_ * 4;
  float* img1      = counts + B_;
  float* img2      = img1 + (size_t)B_ * 3 * H_ * W_;

  nms_kernel<<<B_, 1024, 0, stream>>>(rand_init, scores, num_boxes, ph, pw,
                                      out_boxes, boxes2, counts);
  copy_flip_kernel<<<B_ * 3 * H_, 256, 0, stream>>>(img, img1, img2);
}